// DeepPhenoModel_86139864089465
// MI455X (gfx1250) — compile-verified
//
#include <hip/hip_runtime.h>
#include <math.h>
#include <stdint.h>

// ---------------------------------------------------------------------------
// DeepPheno forward, MI455X (gfx1250, wave32).
//
// Memory-bound (~92 MB, 2.3 GFLOP fp32): W1 (60MB) must stream once at full
// bandwidth. fp32 WMMA 16x16x4 keeps exact precision. HPO maxpool collapses
// to flat[b,j]*colmax[j] (sigmoid>0, hpo>=0).
//
//  * TENSOR_LOAD_TO_LDS (TDM) stages A(64x32)/B(32x32) K-chunks; the D#
//    descriptor does the strided 2D copy, K/N edge zero-fill (tensor_dim =
//    valid extent) and bank-conflict-free LDS padding (A stride 36, B 40).
//  * Double buffer via s_wait_tensorcnt + workgroup barrier: chunk ch+1's
//    DMA overlaps chunk ch's 8 WMMAs.
//  * Deterministic split-K (slabs + fixed-order reduce): GEMM1 47x8 blocks,
//    GEMM2 64x4.
//  * colmax is a two-pass 32-way row-split reduction (the single-pass version
//    was latency-bound: only 2048 threads for a 16.8MB stream).
// ---------------------------------------------------------------------------

typedef __attribute__((ext_vector_type(2))) float v2f;
typedef __attribute__((ext_vector_type(8))) float v8f;
typedef __attribute__((ext_vector_type(4))) unsigned u32x4;
typedef __attribute__((ext_vector_type(4))) int i32x4;
typedef __attribute__((ext_vector_type(8))) int i32x8;

#define B_ROWS 64
#define K1     10000
#define HID    1500
#define EXPW   53
#define K2V    1553   // valid rows of W2
#define K2P    1600   // padded x width (50 * 32)
#define NCLS   2048
#define NP1    1504   // padded GEMM1 N (47 * 32)
#define S1     8      // split-K factor, GEMM1
#define S2     4      // split-K factor, GEMM2
#define KC     32     // K chunk
#define RSPL   32     // colmax row splits (64 rows each)

#if __has_builtin(__builtin_amdgcn_s_wait_tensorcnt)
#define WAIT_TENSORCNT(N) __builtin_amdgcn_s_wait_tensorcnt(N)
#else
#define WAIT_TENSORCNT(N) asm volatile("s_wait_tensorcnt " #N ::: "memory")
#endif

static __device__ __forceinline__ unsigned ldsoff(void* p) {
  return (unsigned)(uintptr_t)(__attribute__((address_space(3))) void*)p;
}

// ---- TDM: 2D tile (4-byte elements) global -> LDS, with LDS row padding ---
// tdim0/tdim1: valid tensor extent from tile start (OOB reads return zero).
// pad codes per D# group1: interval 4 => 32 DWORDs, amount a => a+1 DWORDs.
static __device__ __forceinline__ void tdm_load_2d(
    unsigned lds_byte_off, const void* gaddr, unsigned tdim0, unsigned tdim1,
    unsigned tile0, unsigned tile1, unsigned long long stride0,
    unsigned pad_amount_code) {
  unsigned long long ga = (unsigned long long)(uintptr_t)gaddr;
  u32x4 g0;
  g0[0] = 1u;                                     // count=1, user descriptor
  g0[1] = lds_byte_off;                           // lds_addr
  g0[2] = (unsigned)ga;                           // global_addr[31:0]
  g0[3] = (unsigned)((ga >> 32) & 0x01FFFFFFu)    // global_addr[56:32]
          | (2u << 30);                           // type=2 ("image")
  i32x8 g1;
  g1[0] = (int)((2u << 16)                        // data_size = 4 bytes
                | (1u << 20)                      // pad_enable
                | (4u << 22)                      // pad_interval: 32 DWORDs
                | (pad_amount_code << 25));       // pad_amount
  g1[1] = (int)((tdim0 & 0xFFFFu) << 16);                         // dim0[15:0]
  g1[2] = (int)(((tdim0 >> 16) & 0xFFFFu) | ((tdim1 & 0xFFFFu) << 16));
  g1[3] = (int)(((tdim1 >> 16) & 0xFFFFu) | (tile0 << 16));       // tile_dim0
  g1[4] = (int)(tile1 & 0xFFFFu);                 // tile_dim1 (tile_dim2=0)
  g1[5] = (int)(unsigned)(stride0 & 0xFFFFFFFFull);      // dim0_stride[31:0]
  g1[6] = (int)(unsigned)((stride0 >> 32) & 0xFFFFull);  // dim0_stride[47:32]
  g1[7] = 0;
  i32x4 z4 = {0, 0, 0, 0};
#if __clang_major__ >= 23
  i32x8 z8 = {0, 0, 0, 0, 0, 0, 0, 0};
  __builtin_amdgcn_tensor_load_to_lds(g0, g1, z4, z4, z8, 0);
#else
  __builtin_amdgcn_tensor_load_to_lds(g0, g1, z4, z4, 0);
#endif
}

// --------------- colmax over HPO: two-pass row-split reduction -------------
__global__ __launch_bounds__(256) void colmax_part_kernel(
    const float* __restrict__ hpo, float* __restrict__ pmax) {
  const int j = blockIdx.x * 256 + threadIdx.x;  // column
  const int s = blockIdx.y;                      // row split (64 rows)
  const float* p = hpo + (size_t)s * 64 * NCLS + j;
  float m = p[0];
#pragma unroll 8
  for (int i = 1; i < 64; ++i) m = fmaxf(m, p[(size_t)i * NCLS]);
  pmax[s * NCLS + j] = m;
}

__global__ __launch_bounds__(256) void colmax_final_kernel(
    const float* __restrict__ pmax, float* __restrict__ colmax) {
  const int j = blockIdx.x * 256 + threadIdx.x;
  float m = pmax[j];
#pragma unroll
  for (int s = 1; s < RSPL; ++s) m = fmaxf(m, pmax[s * NCLS + j]);
  colmax[j] = m;
}

// ------------------- x[:, HID:K2P] = [exp_x | zeros] -----------------------
__global__ __launch_bounds__(256) void fill_tail_kernel(
    const float* __restrict__ exp_x, float* __restrict__ x) {
  const int ncols = K2P - HID;  // 100
  int t = blockIdx.x * blockDim.x + threadIdx.x;
  if (t >= B_ROWS * ncols) return;
  int b = t / ncols, j = t - b * ncols;
  x[(size_t)b * K2P + HID + j] = (j < EXPW) ? exp_x[b * EXPW + j] : 0.0f;
}

// ------------------------- split-K WMMA GEMM -------------------------------
// P[split] (64 x nPad) = A(64 x kA) * Bm(kB x nValid) over this split's K
// range.  8 waves -> 64x32 tile; TDM stages A(64x32) and B(32x32) chunks.
__global__ __launch_bounds__(256) void wmma_gemm_splitk(
    const float* __restrict__ A, int lda, int kA,
    const float* __restrict__ Bm, int ldb, int kB, int nValid,
    float* __restrict__ P, int nPad, int cps, int totalChunks) {
  // A stride 36: 36r%64 distinct & 4-aligned for r=0..15 -> the 32-lane b64
  // fragment read covers all 64 banks exactly once.  B stride 40: row pair
  // (k,k+2) reads land in disjoint bank ranges (delta 320B = 80 DWORDs = 16 banks mod 64).
  __shared__ float Al[2][B_ROWS][36];
  __shared__ float Bl[2][KC][40];

  const int tid  = threadIdx.x;
  const int lane = tid & 31;
  const int wave = tid >> 5;
  const int m_base = (wave & 3) * 16;   // 0,16,32,48
  const int n_loc  = (wave >> 2) * 16;  // 0 or 16
  const int n0 = blockIdx.x * 32;
  const int split = blockIdx.y;
  const int cBeg = split * cps;
  const int cEnd = min(cBeg + cps, totalChunks);

  // fragment coords (ISA 7.12.2, 32-bit 16x4 A / 4x16 B / 16x16 C):
  const int mrow = m_base + (lane & 15);
  const int koff = (lane >> 4) * 2;  // lanes 0-15 -> K+0/1, 16-31 -> K+2/3
  const int ncol = n_loc + (lane & 15);

  unsigned alOff[2] = {ldsoff(&Al[0][0][0]), ldsoff(&Al[1][0][0])};
  unsigned blOff[2] = {ldsoff(&Bl[0][0][0]), ldsoff(&Bl[1][0][0])};
  const unsigned nRem = (unsigned)(nValid - n0);  // >=1; <32 only in edge blk

  v8f acc = {};

  // Issue both tile DMAs for chunk ch into buffer buf (wave 0 only).
  auto stage = [&](int buf, int ch) {
    const int k0 = ch * KC;
    int ra = kA - k0;  // valid K for A (tail -> TDM zero-fills cols)
    int rb = kB - k0;  // valid K rows for B
    tdm_load_2d(alOff[buf], A + k0, ra > 0 ? (unsigned)ra : 0u, B_ROWS, KC,
                B_ROWS, (unsigned long long)lda, /*pad 4 DW ->*/ 3u);
    tdm_load_2d(blOff[buf], Bm + (size_t)k0 * ldb + n0, nRem,
                rb > 0 ? (unsigned)rb : 0u, KC, KC, (unsigned long long)ldb,
                /*pad 8 DW ->*/ 7u);
  };

  if (wave == 0) stage(cBeg & 1, cBeg);  // preload first chunk

  for (int ch = cBeg; ch < cEnd; ++ch) {
    const int buf = ch & 1;
    if (wave == 0) {
      if (ch + 1 < cEnd) {
        stage(buf ^ 1, ch + 1);  // 2 TDMs in flight for next chunk
        WAIT_TENSORCNT(2);       // ... current chunk's 2 TDMs complete
      } else {
        WAIT_TENSORCNT(0);
      }
    }
    __syncthreads();  // publish wave0's TENSORcnt wait to all waves

#pragma unroll
    for (int kk = 0; kk < KC; kk += 4) {
      v2f a, b;
      a.x = Al[buf][mrow][kk + koff];
      a.y = Al[buf][mrow][kk + koff + 1];
      b.x = Bl[buf][kk + koff][ncol];
      b.y = Bl[buf][kk + koff + 1][ncol];
      acc = __builtin_amdgcn_wmma_f32_16x16x4_f32(
          /*neg_a=*/false, a, /*neg_b=*/false, b,
          /*c_mod=*/(short)0, acc, /*reuse_a=*/false, /*reuse_b=*/false);
    }
    __syncthreads();  // all reads of buf done before wave0 re-stages it
  }

  // Partial tile -> slab.  C layout: M = m_base + (lane>>4)*8 + g, N = lane&15.
  float* Pt = P + (size_t)split * B_ROWS * nPad;
  const int gn = n0 + ncol;
#pragma unroll
  for (int g = 0; g < 8; ++g) {
    int gm = m_base + (lane >> 4) * 8 + g;
    Pt[(size_t)gm * nPad + gn] = acc[g];
  }
}

// ---------------- fixed-order split-K reductions (deterministic) -----------
__global__ __launch_bounds__(256) void reduce_gelu_kernel(
    const float* __restrict__ P, const float* __restrict__ b1,
    float* __restrict__ x) {
  int t = blockIdx.x * 256 + threadIdx.x;
  if (t >= B_ROWS * NP1) return;
  int m = t / NP1, n = t - m * NP1;
  if (n >= HID) return;  // ghost columns of the padded slab
  float v = 0.0f;
#pragma unroll
  for (int s = 0; s < S1; ++s) v += P[(size_t)s * B_ROWS * NP1 + t];
  v += b1[n];
  v = 0.5f * v * (1.0f + erff(v * 0.70710678118654752f));  // exact gelu
  x[(size_t)m * K2P + n] = v;
}

__global__ __launch_bounds__(256) void reduce_sigmoid_kernel(
    const float* __restrict__ P, const float* __restrict__ b2,
    const float* __restrict__ colmax, float* __restrict__ out) {
  int t = blockIdx.x * 256 + threadIdx.x;
  if (t >= B_ROWS * NCLS) return;
  int n = t & (NCLS - 1);
  float v = 0.0f;
#pragma unroll
  for (int s = 0; s < S2; ++s) v += P[(size_t)s * B_ROWS * NCLS + t];
  v += b2[n];
  out[t] = colmax[n] / (1.0f + __expf(-v));  // sigmoid * colmax
}

// ----------------------------- launcher ------------------------------------
extern "C" void kernel_launch(void* const* d_in, const int* in_sizes, int n_in,
                              void* d_out, int out_size, void* d_ws,
                              size_t ws_size, hipStream_t stream) {
  const float* gos   = (const float*)d_in[0];  // (64, 10000)
  const float* exp_x = (const float*)d_in[1];  // (64, 53)
  const float* W1    = (const float*)d_in[2];  // (10000, 1500)
  const float* b1    = (const float*)d_in[3];  // (1500,)
  const float* W2    = (const float*)d_in[4];  // (1553, 2048)
  const float* b2    = (const float*)d_in[5];  // (2048,)
  const float* hpo   = (const float*)d_in[6];  // (2048, 2048)
  float* out = (float*)d_out;                  // (64, 2048)

  // ws layout (floats):
  //   colmax 2048 | pmax 32x2048 | xbuf 64x1600 | P1 8x64x1504 | P2 4x64x2048
  float* colmax = (float*)d_ws;
  float* pmax = colmax + NCLS;
  float* xbuf = pmax + (size_t)RSPL * NCLS;
  float* P1 = xbuf + (size_t)B_ROWS * K2P;
  float* P2 = P1 + (size_t)S1 * B_ROWS * NP1;  // total ~5.9 MB

  // 1) column max of HPO: 256 blocks of partials, then final reduce
  colmax_part_kernel<<<dim3(NCLS / 256, RSPL), 256, 0, stream>>>(hpo, pmax);
  colmax_final_kernel<<<NCLS / 256, 256, 0, stream>>>(pmax, colmax);

  // 2) x[:, 1500:1600] = [exp_x | 0]
  fill_tail_kernel<<<(B_ROWS * (K2P - HID) + 255) / 256, 256, 0, stream>>>(
      exp_x, xbuf);

  // 3) GEMM1 partials: 313 chunks of 32 (TDM zero-fills the 16-wide tail),
  //    split 8 ways -> cps 40.
  wmma_gemm_splitk<<<dim3(NP1 / 32, S1), 256, 0, stream>>>(
      gos, K1, K1, W1, HID, K1, HID, P1, NP1, 40, 313);

  // 4) reduce + bias + gelu -> x[:, 0:1500]
  reduce_gelu_kernel<<<(B_ROWS * NP1 + 255) / 256, 256, 0, stream>>>(P1, b1,
                                                                     xbuf);

  // 5) GEMM2 partials: 50 chunks of 32 (W2 rows >=1553 zero-filled by TDM),
  //    split 4 ways -> cps 13.
  wmma_gemm_splitk<<<dim3(NCLS / 32, S2), 256, 0, stream>>>(
      xbuf, K2P, K2P, W2, NCLS, K2V, NCLS, P2, NCLS, 13, 50);

  // 6) reduce + bias + sigmoid*colmax -> out
  reduce_sigmoid_kernel<<<(B_ROWS * NCLS + 255) / 256, 256, 0, stream>>>(
      P2, b2, colmax, out);
}